// GCN_40321152975136
// MI455X (gfx1250) — compile-verified
//
#include <hip/hip_runtime.h>

#define F 128          // feature width for all GCN layers
#define WAVE 32
#define PSTRIDE 288    // LDS stride per K-pair (256 data + 32 pad) -> conflict-free b64 reads

typedef __attribute__((ext_vector_type(2))) float v2f;
typedef __attribute__((ext_vector_type(8))) float v8f;

__device__ __forceinline__ void atomAddF(float* p, float v) {
  __hip_atomic_fetch_add(p, v, __ATOMIC_RELAXED, __HIP_MEMORY_SCOPE_AGENT);
}

// ---------------- utility ----------------
__global__ void k_fill(float* __restrict__ p, float v, int n) {
  int i = blockIdx.x * blockDim.x + threadIdx.x;
  if (i < n) p[i] = v;
}

// ---------------- degree / dinv ----------------
__global__ void k_deg_edges(const int* __restrict__ dstIdx, float* __restrict__ deg, int nE) {
  int e = blockIdx.x * blockDim.x + threadIdx.x;
  if (e < nE) atomAddF(&deg[dstIdx[e]], 1.0f);
}

__global__ void k_deg_to_dinv(float* __restrict__ deg, int n) {
  int i = blockIdx.x * blockDim.x + threadIdx.x;
  if (i < n) {
    float d = deg[i];
    deg[i] = (d > 0.0f) ? rsqrtf(fmaxf(d, 1.0f)) : 0.0f;
  }
}

// ---------------- dense GEMM: H[n x 128] = X[n x 128] @ W[128 x 128] ----------------
// fp32 WMMA 16x16x4. Block = 128 threads = 4 waves; each wave owns TWO 16-row tiles
// sharing the same B fragments, so each ds_load_b64 feeds 2 WMMAs (halves exposed
// LDS latency per matrix op). Occupancy is LDS-capped (72KB -> 4 blocks/WGP), so
// __launch_bounds__(128, 1) lifts the VGPR budget to fit the 128 accumulator
// registers without spilling — register headroom is free here.
// W staged in LDS K-pair-interleaved (sW[p][n][q] = W[2p+q][n]) so each lane's
// B fragment is one aligned ds_load_b64; p-stride padded to 288 floats so the two
// lane-halves (p vs p+1) hit disjoint 32-bank groups (conflict-free).
__global__ void __launch_bounds__(128, 1)
k_gemm_xw(const float* __restrict__ X, const float* __restrict__ W,
          float* __restrict__ H, int nRows) {
  __shared__ float sW[64 * PSTRIDE];          // 72 KB
  const int tid  = threadIdx.x;               // 0..127
  const int wave = tid >> 5;                  // 0..3
  const int lane = tid & 31;

  // cooperative pack of W into LDS: idx -> (p = K-pair, n = column)
  for (int idx = tid; idx < 64 * F; idx += 128) {
    const int p = idx >> 7;
    const int n = idx & (F - 1);
    v2f w;
    w.x = W[(2 * p)     * F + n];             // coalesced along n
    w.y = W[(2 * p + 1) * F + n];
    *(v2f*)&sW[p * PSTRIDE + n * 2] = w;      // 8B store
  }
  __syncthreads();

  const int tile0 = blockIdx.x * 8 + wave * 2;  // first of two 16-row tiles
  if (tile0 * 16 >= nRows) return;              // whole-wave exit (EXEC all-1 for WMMA)
  const bool tile1ok = (tile0 + 2) * 16 <= nRows;

  const int m  = lane & 15;                   // row within A tile / col within B tile
  const int kg = lane >> 4;                   // lane-half -> K subgroup {0,1}
  const int r0 = tile0 * 16 + m;
  int r1 = (tile0 + 1) * 16 + m;
  if (r1 >= nRows) r1 = nRows - 1;            // clamp (redundant work, uniform EXEC)
  const float* __restrict__ xrow0 = X + (size_t)r0 * F;
  const float* __restrict__ xrow1 = X + (size_t)r1 * F;

  v8f acc0[8], acc1[8];
#pragma unroll
  for (int t = 0; t < 8; ++t) { acc0[t] = {}; acc1[t] = {}; }

  for (int kk = 0; kk < F; kk += 4) {
    const int ka = kk + 2 * kg;               // this lane-half's K base (always even)
    const int p  = (kk >> 1) + kg;            // K-pair index in LDS
    const v2f a0 = *(const v2f*)&xrow0[ka];   // single global_load_b64 each
    const v2f a1 = *(const v2f*)&xrow1[ka];
    const float* __restrict__ brow = &sW[p * PSTRIDE + m * 2];
#pragma unroll
    for (int t = 0; t < 8; ++t) {             // each B fragment feeds 2 WMMAs
      const v2f b = *(const v2f*)&brow[t * 32];
      acc0[t] = __builtin_amdgcn_wmma_f32_16x16x4_f32(
          false, a0, false, b, (short)0, acc0[t], false, false);
      acc1[t] = __builtin_amdgcn_wmma_f32_16x16x4_f32(
          false, a1, false, b, (short)0, acc1[t], false, false);
    }
  }

  // C/D layout: VGPR r -> row r (lanes 0-15) or row r+8 (lanes 16-31), col = lane&15
  const int rb0 = tile0 * 16 + (lane >> 4) * 8;
#pragma unroll
  for (int t = 0; t < 8; ++t) {
#pragma unroll
    for (int r = 0; r < 8; ++r) {
      H[(size_t)(rb0 + r) * F + t * 16 + m] = acc0[t][r];
    }
  }
  if (tile1ok) {
    const int rb1 = rb0 + 16;
#pragma unroll
    for (int t = 0; t < 8; ++t) {
#pragma unroll
      for (int r = 0; r < 8; ++r) {
        H[(size_t)(rb1 + r) * F + t * 16 + m] = acc1[t][r];
      }
    }
  }
}

// ---------------- self-loop + bias init: out[i] = h[i]*dinv[i]^2 + b ----------------
__global__ void k_self_loop_bias(const float* __restrict__ h, const float* __restrict__ dinv,
                                 const float* __restrict__ b, float* __restrict__ out) {
  const int i = blockIdx.x;       // node
  const int f = threadIdx.x;      // feature (128 threads)
  const float dv = dinv[i];
  out[(size_t)i * F + f] = h[(size_t)i * F + f] * dv * dv + b[f];
}

// ---------------- edge scatter: out[dst] += h[src] * dinv[src]*dinv[dst] ----------------
// one wave32 per edge, float4 per lane (coalesced 512B row), hardware f32 atomics
__global__ void k_edge_scatter(const float* __restrict__ h, const float* __restrict__ dinv,
                               const int* __restrict__ srcIdx, const int* __restrict__ dstIdx,
                               float* __restrict__ out, int nE) {
  const int e = (blockIdx.x * blockDim.x + threadIdx.x) >> 5;
  const int lane = threadIdx.x & 31;
  if (e >= nE) return;
  const int s = srcIdx[e];
  const int d = dstIdx[e];
  const float nrm = dinv[s] * dinv[d];
  const float4 hv = *(const float4*)&h[(size_t)s * F + lane * 4];
  float* o = &out[(size_t)d * F + lane * 4];
  atomAddF(o + 0, hv.x * nrm);
  atomAddF(o + 1, hv.y * nrm);
  atomAddF(o + 2, hv.z * nrm);
  atomAddF(o + 3, hv.w * nrm);
}

// ---------------- relu ----------------
__global__ void k_relu(float* __restrict__ x, int n) {
  int i = blockIdx.x * blockDim.x + threadIdx.x;
  if (i < n) x[i] = fmaxf(x[i], 0.0f);
}

// ---------------- global mean pool ----------------
__global__ void k_pool_sum(const float* __restrict__ h, const int* __restrict__ batch,
                           float* __restrict__ sums, float* __restrict__ cnt) {
  const int i = blockIdx.x;       // node
  const int f = threadIdx.x;      // feature
  const int g = batch[i];
  atomAddF(&sums[(size_t)g * F + f], h[(size_t)i * F + f]);
  if (f == 0) atomAddF(&cnt[g], 1.0f);
}

__global__ void k_pool_div(float* __restrict__ sums, const float* __restrict__ cnt) {
  const int g = blockIdx.x;
  const int f = threadIdx.x;
  sums[(size_t)g * F + f] /= fmaxf(cnt[g], 1.0f);
}

// ---------------- MLP head ----------------
__global__ void k_fc1(const float* __restrict__ hg, const float* __restrict__ Wf1,
                      const float* __restrict__ bf1, float* __restrict__ o, int Hf) {
  const int g = blockIdx.x;
  const int j = threadIdx.x;      // Hf threads (64)
  float acc = bf1[j];
  for (int k = 0; k < F; ++k) acc += hg[(size_t)g * F + k] * Wf1[k * Hf + j];
  o[(size_t)g * Hf + j] = fmaxf(acc, 0.0f);
}

__global__ void k_fc2(const float* __restrict__ f1, const float* __restrict__ Wf2,
                      const float* __restrict__ bf2, float* __restrict__ out, int G, int Hf) {
  const int g = blockIdx.x * blockDim.x + threadIdx.x;
  if (g >= G) return;
  float acc = bf2[0];
  for (int k = 0; k < Hf; ++k) acc += f1[(size_t)g * Hf + k] * Wf2[k];
  out[g] = acc;
}

extern "C" void kernel_launch(void* const* d_in, const int* in_sizes, int n_in,
                              void* d_out, int out_size, void* d_ws, size_t ws_size,
                              hipStream_t stream) {
  const float* x     = (const float*)d_in[0];
  const int*   eidx  = (const int*)d_in[1];
  const int*   batch = (const int*)d_in[2];
  const float* W1 = (const float*)d_in[3];  const float* b1 = (const float*)d_in[4];
  const float* W2 = (const float*)d_in[5];  const float* b2 = (const float*)d_in[6];
  const float* W3 = (const float*)d_in[7];  const float* b3 = (const float*)d_in[8];
  const float* Wf1 = (const float*)d_in[9]; const float* bf1 = (const float*)d_in[10];
  const float* Wf2 = (const float*)d_in[11]; const float* bf2 = (const float*)d_in[12];

  const int n  = in_sizes[0] / F;     // 50000 nodes
  const int E  = in_sizes[1] / 2;     // 600000 edges
  const int G  = out_size;            // 500 graphs (D_out = 1)
  const int Hf = in_sizes[10];        // 64

  const int* srcIdx = eidx;
  const int* dstIdx = eidx + E;

  // workspace layout
  float* hbuf = (float*)d_ws;             // n*F
  float* abuf = hbuf + (size_t)n * F;     // n*F
  float* dinv = abuf + (size_t)n * F;     // n     (deg -> dinv in place)
  float* sums = dinv + n;                 // G*F
  float* cnt  = sums + (size_t)G * F;     // G
  float* f1   = cnt + G;                  // G*Hf

  const int nf = n * F;

  // ---- degree / dinv (once) ----
  k_fill<<<(n + 255) / 256, 256, 0, stream>>>(dinv, 1.0f, n);              // self-loop
  k_deg_edges<<<(E + 255) / 256, 256, 0, stream>>>(dstIdx, dinv, E);
  k_deg_to_dinv<<<(n + 255) / 256, 256, 0, stream>>>(dinv, n);

  const int tiles = (n + 15) / 16;                    // 16-row tiles
  const int gemmBlocks = (tiles + 7) / 8;             // 2 tiles/wave, 4 waves/block
  const int scatBlocks = (E * WAVE + 255) / 256;      // one wave per edge

  // ---- 3 GCN layers ----
  const float* layerIn[3] = { x, abuf, abuf };
  const float* Ws[3] = { W1, W2, W3 };
  const float* bs[3] = { b1, b2, b3 };
  for (int l = 0; l < 3; ++l) {
    k_gemm_xw<<<gemmBlocks, 128, 0, stream>>>(layerIn[l], Ws[l], hbuf, n);
    k_self_loop_bias<<<n, F, 0, stream>>>(hbuf, dinv, bs[l], abuf);
    k_edge_scatter<<<scatBlocks, 256, 0, stream>>>(hbuf, dinv, srcIdx, dstIdx, abuf, E);
    k_relu<<<(nf + 255) / 256, 256, 0, stream>>>(abuf, nf);
  }

  // ---- global mean pool ----
  k_fill<<<(G * F + G + 255) / 256, 256, 0, stream>>>(sums, 0.0f, G * F + G); // sums + cnt
  k_pool_sum<<<n, F, 0, stream>>>(abuf, batch, sums, cnt);
  k_pool_div<<<G, F, 0, stream>>>(sums, cnt);

  // ---- MLP head ----
  k_fc1<<<G, Hf, 0, stream>>>(sums, Wf1, bf1, f1, Hf);
  k_fc2<<<(G + 63) / 64, 64, 0, stream>>>(f1, Wf2, bf2, (float*)d_out, G, Hf);
}